// LFPCompetitionLoss_16896401342589
// MI455X (gfx1250) — compile-verified
//
#include <hip/hip_runtime.h>
#include <hip/hip_bf16.h>
#include <math.h>

#define GAMMA_ 0.1f
#define ALPHA_ 0.3f
#define EPS_   1e-8f
#define BIG_   1e8f
#define NN     512
#define BB     256
#define BDIM   256

// d_ws layout (floats):
//  [0*BB, 1*BB): soft-DTW terminal value per batch
//  [1*BB, 2*BB): sum(p)      per batch
//  [2*BB, 3*BB): sum(t)      per batch
//  [3*BB, 4*BB): sum(p*t)    per batch   (WMMA Gram diag)
//  [4*BB, 5*BB): sum(p*p)    per batch   (WMMA Gram diag)
//  [5*BB, 6*BB): sum(t*t)    per batch   (WMMA Gram diag)

#if defined(__gfx1250__) && __has_builtin(__builtin_amdgcn_global_load_async_to_lds_b64) && __has_builtin(__builtin_amdgcn_s_wait_asynccnt)
#define USE_ASYNC_LDS 1
#else
#define USE_ASYNC_LDS 0
#endif

#if defined(__gfx1250__) && __has_builtin(__builtin_amdgcn_wmma_f32_16x16x4_f32)
#define USE_WMMA_F32 1
#else
#define USE_WMMA_F32 0
#endif

typedef float v2f __attribute__((ext_vector_type(2)));
typedef float v8f __attribute__((ext_vector_type(8)));
typedef int   v2i __attribute__((vector_size(8)));

// ---------------------------------------------------------------------------
// Kernel 1: per-batch soft-DTW wavefront + row sums.
// One workgroup per batch row. Rows + 3 rolling diagonals live in LDS.
// ---------------------------------------------------------------------------
__global__ __launch_bounds__(BDIM) void dtw_rowsum_kernel(
    const float* __restrict__ pred, const float* __restrict__ target,
    float* __restrict__ ws) {
  __shared__ float s_pred[NN];
  __shared__ float s_targ[NN];
  __shared__ float diag[3][NN + 1];
  __shared__ float redp[BDIM / 32];
  __shared__ float redt[BDIM / 32];

  const int b   = blockIdx.x;
  const int tid = threadIdx.x;
  const float* prow = pred + b * NN;
  const float* trow = target + b * NN;

  // ---- Stage the two rows into LDS (async DMA path if the toolchain has it)
#if USE_ASYNC_LDS
  {
    typedef __attribute__((address_space(1))) v2i GV2;
    typedef __attribute__((address_space(3))) v2i LV2;
    __builtin_amdgcn_global_load_async_to_lds_b64(
        (GV2*)(prow + 2 * tid), (LV2*)(&s_pred[2 * tid]), 0, 0);
    __builtin_amdgcn_global_load_async_to_lds_b64(
        (GV2*)(trow + 2 * tid), (LV2*)(&s_targ[2 * tid]), 0, 0);
    __builtin_amdgcn_s_wait_asynccnt(0);
  }
#else
  for (int i = tid; i < NN; i += BDIM) {
    s_pred[i] = prow[i];
    s_targ[i] = trow[i];
  }
#endif
  __syncthreads();

  // ---- Row sums for the Pearson term (wave shuffle + cross-wave LDS).
  float sum_p = 0.f, sum_t = 0.f;
  for (int i = tid; i < NN; i += BDIM) {
    sum_p += s_pred[i];
    sum_t += s_targ[i];
  }
  for (int off = 16; off > 0; off >>= 1) {
    sum_p += __shfl_xor(sum_p, off);
    sum_t += __shfl_xor(sum_t, off);
  }
  const int wave = tid >> 5;
  if ((tid & 31) == 0) { redp[wave] = sum_p; redt[wave] = sum_t; }

  // ---- Init rolling diagonals: d=0 (a0) and d=1 (a1).
  for (int i = tid; i <= NN; i += BDIM) {
    diag[0][i] = (i == 0) ? 0.0f : BIG_;
    diag[1][i] = BIG_;
    diag[2][i] = BIG_;
  }
  __syncthreads();

  if (tid == 0) {
    float ap = 0.f, at = 0.f;
    for (int w = 0; w < BDIM / 32; ++w) { ap += redp[w]; at += redt[w]; }
    ws[1 * BB + b] = ap;
    ws[2 * BB + b] = at;
  }

  // ---- Anti-diagonal wavefront, 1023 steps, one barrier per step.
  const float inv_g = 1.0f / GAMMA_;
  for (int d = 2; d <= 2 * NN; ++d) {
    float*       cur = diag[d % 3];
    const float* dm1 = diag[(d + 2) % 3];  // d-1
    const float* dm2 = diag[(d + 1) % 3];  // d-2
    for (int i = tid; i <= NN; i += BDIM) {
      const int j = d - i;
      float out = BIG_;
      if (i >= 1 && j >= 1 && j <= NN) {
        const float x = dm1[i - 1];
        const float y = dm1[i];
        const float z = dm2[i - 1];
        float m = fminf(x, fminf(y, z));
        // softmin = m - gamma * log( sum exp((m - v)/gamma) )  (TRANS pipe)
        float s = __expf((m - x) * inv_g) + __expf((m - y) * inv_g) +
                  __expf((m - z) * inv_g);
        float softmin = m - GAMMA_ * __logf(s);
        float dp = s_pred[i - 1] - s_targ[j - 1];
        out = dp * dp + softmin;
      }
      cur[i] = out;
    }
    __syncthreads();
  }

  if (tid == 0) ws[b] = diag[(2 * NN) % 3][NN];
}

// ---------------------------------------------------------------------------
// Kernel 2: Gram diagonals via V_WMMA_F32_16X16X4_F32 (f32 WMMA, full fidelity).
// One wave per 16 batch rows; EXEC all-ones as WMMA requires.
// C[m][n] = sum_k A[m][k] * B[k][n]; we accumulate P·T^T, P·P^T, T·T^T over
// K-chunks of 4 and extract the diagonal.
// ---------------------------------------------------------------------------
__global__ __launch_bounds__(32) void gram_kernel(
    const float* __restrict__ pred, const float* __restrict__ target,
    float* __restrict__ ws) {
  const int g    = blockIdx.x;      // 16 groups of 16 batches
  const int lane = threadIdx.x;     // 32 lanes, no divergence before WMMA
  const int m0   = g * 16;
  const int row  = lane & 15;
  const int half = lane >> 4;

  const float* pr = pred + (m0 + row) * NN;
  const float* tr = target + (m0 + row) * NN;

#if USE_WMMA_F32
  v8f cPT = {}; v8f cPP = {}; v8f cTT = {};
  // A 16x4 f32 layout: lanes 0-15 hold M=lane, K={k0,k0+1}; lanes 16-31 hold
  // M=lane-16, K={k0+2,k0+3}. B 4x16 mirrors the striping with N=lane%16.
  for (int k0 = 0; k0 < NN; k0 += 4) {
    const int kb = k0 + 2 * half;
    v2f a;  a.x  = pr[kb]; a.y  = pr[kb + 1];
    v2f bt; bt.x = tr[kb]; bt.y = tr[kb + 1];
    cPT = __builtin_amdgcn_wmma_f32_16x16x4_f32(false, a, false, bt,
                                                (short)0, cPT, false, false);
    cPP = __builtin_amdgcn_wmma_f32_16x16x4_f32(false, a, false, a,
                                                (short)0, cPP, false, false);
    cTT = __builtin_amdgcn_wmma_f32_16x16x4_f32(false, bt, false, bt,
                                                (short)0, cTT, false, false);
  }
  // C/D layout: VGPR r -> lanes 0-15: (M=r, N=lane); lanes 16-31: (M=r+8, N=lane-16).
  // Diagonal (M==N): VGPR r at lane r (low half) and lane r+24 (high half).
#pragma unroll
  for (int r = 0; r < 8; ++r) {
    if (lane == r) {
      ws[3 * BB + m0 + r] = cPT[r];
      ws[4 * BB + m0 + r] = cPP[r];
      ws[5 * BB + m0 + r] = cTT[r];
    }
    if (lane == r + 24) {
      ws[3 * BB + m0 + r + 8] = cPT[r];
      ws[4 * BB + m0 + r + 8] = cPP[r];
      ws[5 * BB + m0 + r + 8] = cTT[r];
    }
  }
#else
  // Scalar fallback: two lanes split K for each of the 16 rows.
  float spt = 0.f, spp = 0.f, stt = 0.f;
  const int k0 = half * (NN / 2);
  for (int k = k0; k < k0 + NN / 2; ++k) {
    const float p = pr[k], t = tr[k];
    spt += p * t; spp += p * p; stt += t * t;
  }
  spt += __shfl_xor(spt, 16);
  spp += __shfl_xor(spp, 16);
  stt += __shfl_xor(stt, 16);
  if (lane < 16) {
    ws[3 * BB + m0 + row] = spt;
    ws[4 * BB + m0 + row] = spp;
    ws[5 * BB + m0 + row] = stt;
  }
#endif
}

// ---------------------------------------------------------------------------
// Kernel 3: combine 256 per-batch partials into the scalar loss.
// ---------------------------------------------------------------------------
__global__ __launch_bounds__(BDIM) void finalize_kernel(
    const float* __restrict__ ws, float* __restrict__ out) {
  __shared__ float redd[BDIM / 32];
  __shared__ float redr[BDIM / 32];
  const int t = threadIdx.x;  // 256 threads == 256 batches

  const float dtw = ws[t];
  const float sp  = ws[1 * BB + t];
  const float st  = ws[2 * BB + t];
  const float spt = ws[3 * BB + t];
  const float spp = ws[4 * BB + t];
  const float stt = ws[5 * BB + t];

  const float invN = 1.0f / (float)NN;
  const float dot = spt - sp * st * invN;
  const float np  = sqrtf(fmaxf(spp - sp * sp * invN, 0.0f));
  const float nt  = sqrtf(fmaxf(stt - st * st * invN, 0.0f));
  float pear = dot / fmaxf(np * nt, EPS_);

  float d = dtw, r = pear;
  for (int off = 16; off > 0; off >>= 1) {
    d += __shfl_xor(d, off);
    r += __shfl_xor(r, off);
  }
  const int wave = t >> 5;
  if ((t & 31) == 0) { redd[wave] = d; redr[wave] = r; }
  __syncthreads();
  if (t == 0) {
    float sd = 0.f, sr = 0.f;
    for (int w = 0; w < BDIM / 32; ++w) { sd += redd[w]; sr += redr[w]; }
    const float dtw_mean     = sd / (float)BB;
    const float pearson_loss = 1.0f - sr / (float)BB;
    out[0] = ALPHA_ * dtw_mean + (1.0f - ALPHA_) * pearson_loss;
  }
}

extern "C" void kernel_launch(void* const* d_in, const int* in_sizes, int n_in,
                              void* d_out, int out_size, void* d_ws, size_t ws_size,
                              hipStream_t stream) {
  const float* pred   = (const float*)d_in[0];
  const float* target = (const float*)d_in[1];
  float* ws  = (float*)d_ws;
  float* out = (float*)d_out;

  dtw_rowsum_kernel<<<BB, BDIM, 0, stream>>>(pred, target, ws);
  gram_kernel<<<BB / 16, 32, 0, stream>>>(pred, target, ws);
  finalize_kernel<<<1, BDIM, 0, stream>>>(ws, out);
}